// TGDiffusion_46359876993479
// MI455X (gfx1250) — compile-verified
//
#include <hip/hip_runtime.h>
#include <stdint.h>

// Problem constants (match the reference).
static constexpr int B  = 512;
static constexpr int P  = 4;
static constexpr int A  = 32;
static constexpr int T  = 8;
static constexpr int NK = 21;          // K = -10..10

static constexpr int TPB   = 256;              // 8 wave32 per block
static constexpr int ELEMS = T * P * A * 3;    // 3072 elements per graph b
static constexpr int EPT   = ELEMS / TPB;      // 12 elements per thread
static constexpr int GRP   = A * 3;            // 96 elements per (t,p) group
static constexpr int NGRP  = T * P;            // 32 groups per b

// Flat LDS-staging word layout per block:
//   [0,96)    x_t        (A*3 f32)
//   [96,480)  perm       (P*A*3 f32)
//   [480,608) helper     (P*A i32)
//   [608,704) shifts     (T*P*3 f32)
static constexpr int W_XT   = 0;
static constexpr int W_PERM = 96;
static constexpr int W_HELP = 480;
static constexpr int W_SHFT = 608;
static constexpr int W_TOT  = 704;

__device__ __forceinline__ uint32_t lds_off(const void* p) {
  // Generic (flat) pointers to LDS carry the aperture tag in the high dword;
  // addr[31:0] is the LDS byte offset (ISA 10.2 aperture mapping).
  return (uint32_t)(uintptr_t)p;
}

__device__ __forceinline__ void async_load_dword_to_lds(uint32_t lds_byte_off,
                                                        const void* gaddr) {
  uint64_t ga = (uint64_t)(uintptr_t)gaddr;
  // CDNA5 async LDS DMA: LDS byte address comes from the VDST VGPR; tracked by ASYNCcnt.
  asm volatile("global_load_async_to_lds_b32 %0, %1, off"
               :
               : "v"(lds_byte_off), "v"(ga)
               : "memory");
}

__global__ __launch_bounds__(TPB) void tg_diffusion_kernel(
    const float* __restrict__ frac_coords_t,     // (B*A, 3)
    const float* __restrict__ permuted_frac,     // (B*P*A, 3)
    const float* __restrict__ sigmas,            // (B,)
    const float* __restrict__ random_shifts,     // (T, B*P, 3)
    const int*   __restrict__ helper_idx,        // (B*P*A,)
    float*       __restrict__ out)               // (B*A, 3)
{
  __shared__ float s_xt  [A * 3];        // 96
  __shared__ float s_perm[P * A * 3];    // 384
  __shared__ int   s_help[P * A];        // 128
  __shared__ float s_shft[T * P * 3];    // 96
  __shared__ float s_L   [NGRP];         // per-(t,p) log_p_graph
  __shared__ float s_W   [NGRP];         // softmax weights
  __shared__ float s_out [A * 3];        // segment-summed output rows

  const int b   = blockIdx.x;
  const int tid = threadIdx.x;

  // ---- Stage all per-b inputs into LDS with gfx1250 async LDS DMA ----------
  for (int w = tid; w < W_TOT; w += TPB) {
    uint32_t la;
    const void* ga;
    if (w < W_PERM) {                          // x_t
      int i = w - W_XT;
      ga = frac_coords_t + (size_t)b * (A * 3) + i;
      la = lds_off(&s_xt[i]);
    } else if (w < W_HELP) {                   // perm
      int i = w - W_PERM;
      ga = permuted_frac + (size_t)b * (P * A * 3) + i;
      la = lds_off(&s_perm[i]);
    } else if (w < W_SHFT) {                   // helper (i32)
      int i = w - W_HELP;
      ga = helper_idx + (size_t)b * (P * A) + i;
      la = lds_off(&s_help[i]);
    } else {                                   // shifts: (T, B*P, 3) strided
      int i = w - W_SHFT;                      // i = t*12 + (p*3 + d)
      int t = i / (P * 3);
      int r = i % (P * 3);
      ga = random_shifts + (size_t)t * (B * P * 3) + (size_t)b * (P * 3) + r;
      la = lds_off(&s_shft[i]);
    }
    async_load_dword_to_lds(la, ga);
  }

  if (tid < NGRP)  s_L[tid]   = 0.0f;
  if (tid < A * 3) s_out[tid] = 0.0f;

  asm volatile("s_wait_asynccnt 0" ::: "memory");
  __syncthreads();

  const float sg     = sigmas[b];
  const float inv2s2 = 1.0f / (2.0f * sg * sg);
  const float invs2  = 1.0f / (sg * sg);

  // ---- Per-element logsumexp over K, group log-p sums, per-element score ---
  float sc[EPT];
  float gsum = 0.0f;
  int   gcur = (tid * EPT) / GRP;

#pragma unroll
  for (int i = 0; i < EPT; ++i) {
    const int e = tid * EPT + i;
    const int d = e % 3;
    const int a = (e / 3) % A;
    const int p = (e / GRP) % P;
    const int t = e / (P * GRP);
    const int g = e / GRP;                    // g = t*P + p (matches reference)

    const float u  = s_perm[(p * A + a) * 3 + d] + s_shft[(t * P + p) * 3 + d] + 1.0f;
    const float ap = u - floorf(u);           // (perm + shift + 1) mod 1
    const float x  = s_xt[a * 3 + d] - ap;

    // logsumexp over k = -10..10 with max subtraction (matches jax numerics)
    float m = -3.402823466e38f;
#pragma unroll
    for (int k = 0; k < NK; ++k) {
      const float dk = (float)(k - 10) - x;
      m = fmaxf(m, -(dk * dk) * inv2s2);
    }
    float ssum = 0.0f, num = 0.0f;
#pragma unroll
    for (int k = 0; k < NK; ++k) {
      const float dk = (float)(k - 10) - x;
      const float ex = __expf(-(dk * dk) * inv2s2 - m);
      ssum += ex;
      num  += ex * dk;
    }

    const float logp = m + __logf(ssum);
    sc[i] = (num / ssum) * invs2;             // Σ_k w_k (k - x) / σ²

    if (g != gcur) {                          // ≤ 2 LDS atomics per thread
      atomicAdd(&s_L[gcur], gsum);
      gsum = 0.0f;
      gcur = g;
    }
    gsum += logp;
  }
  atomicAdd(&s_L[gcur], gsum);
  __syncthreads();

  // ---- Softmax over the 32 (t,p) hypotheses of this graph ------------------
  if (tid < NGRP) {
    float mx = -3.402823466e38f;
#pragma unroll
    for (int j = 0; j < NGRP; ++j) mx = fmaxf(mx, s_L[j]);
    float se = 0.0f;
#pragma unroll
    for (int j = 0; j < NGRP; ++j) se += __expf(s_L[j] - mx);
    s_W[tid] = __expf(s_L[tid] - mx) / se;
  }
  __syncthreads();

  // ---- Weighted score, segment-sum into rows helper[p,a] of this b ---------
#pragma unroll
  for (int i = 0; i < EPT; ++i) {
    const int e = tid * EPT + i;
    const int d = e % 3;
    const int a = (e / 3) % A;
    const int p = (e / GRP) % P;
    const int g = e / GRP;
    const int h = s_help[p * A + a];          // in [0, A)
    atomicAdd(&s_out[h * 3 + d], s_W[g] * sc[i]);
  }
  __syncthreads();

  // ---- Store: each output element written exactly once across the grid ----
  if (tid < A * 3) out[(size_t)b * (A * 3) + tid] = s_out[tid];
}

extern "C" void kernel_launch(void* const* d_in, const int* in_sizes, int n_in,
                              void* d_out, int out_size, void* d_ws, size_t ws_size,
                              hipStream_t stream) {
  (void)in_sizes; (void)n_in; (void)out_size; (void)d_ws; (void)ws_size;
  const float* frac_coords_t = (const float*)d_in[0];
  const float* permuted_frac = (const float*)d_in[1];
  const float* sigmas        = (const float*)d_in[2];
  const float* random_shifts = (const float*)d_in[3];
  const int*   helper_idx    = (const int*)d_in[4];
  float*       out           = (float*)d_out;

  tg_diffusion_kernel<<<dim3(B), dim3(TPB), 0, stream>>>(
      frac_coords_t, permuted_frac, sigmas, random_shifts, helper_idx, out);
}